// MaskedDenseLayer_39359080300845
// MI455X (gfx1250) — compile-verified
//
#include <hip/hip_runtime.h>

// ---------------------------------------------------------------------------
// MaskedDenseLayer for MI455X (gfx1250, wave32).
// out[b,n] = relu( sum_k x[b,k] * W[k,n] * M[state[b],k,n] )
// BS=256, K=1024, N=1024, NUM_STATES=20.
//
// Strategy: counting-sort rows by state (mask id), then per-state tiled GEMM
// with B = W .* M[s] built on the fly. Masks (80 MB total) are the dominant
// traffic; grouping makes them compulsory-read-once and they fit in the
// 192 MB L2, so HBM traffic ~86 MB (~3.7 us at 23.3 TB/s). f32 WMMA
// (V_WMMA_F32_16X16X4_F32) keeps full reference precision; problem is
// memory-bound so its lower K is irrelevant.
// ---------------------------------------------------------------------------

#define NUM_STATES 20
#define BS         256
#define K_IN       1024
#define N_OUT      1024

#define TILE_M       16          // rows per WMMA tile
#define KC           256         // K chunk staged in LDS
#define A_STRIDE     (KC + 4)    // pad: 260 % 64 = 4 banks/row -> conflict free
#define WAVES        4           // 4 waves * 16 cols = 64 cols per block
#define BLOCK_THREADS (WAVES * 32)
#define MAX_ROW_TILES (BS / TILE_M)   // 16 worst-case tiles per state

typedef __attribute__((ext_vector_type(2))) float v2f;
typedef __attribute__((ext_vector_type(4))) float v4f;
typedef __attribute__((ext_vector_type(8))) float v8f;

// ---------------------------------------------------------------------------
// Kernel 1: deterministic stable counting sort of the 256 rows by state.
// ws layout: int sortedRows[256]; int groupStart[20]; int groupCount[20];
// ---------------------------------------------------------------------------
__global__ __launch_bounds__(BS)
void sort_rows_kernel(const int* __restrict__ state,
                      int* __restrict__ sortedRows,
                      int* __restrict__ groupStart,
                      int* __restrict__ groupCount) {
    __shared__ int sh[BS];
    const int t = threadIdx.x;
    sh[t] = state[t];
    __syncthreads();

    const int s = sh[t];
    int start = 0;   // #rows with state < s
    int rank  = 0;   // #rows i < t with state == s  (stable)
    for (int i = 0; i < BS; ++i) {
        const int v = sh[i];
        start += (v < s) ? 1 : 0;
        rank  += (v == s && i < t) ? 1 : 0;
    }
    sortedRows[start + rank] = t;

    if (t < NUM_STATES) {
        int st = 0, c = 0;
        for (int i = 0; i < BS; ++i) {
            const int v = sh[i];
            st += (v < t) ? 1 : 0;
            c  += (v == t) ? 1 : 0;
        }
        groupStart[t] = st;
        groupCount[t] = c;
    }
}

// ---------------------------------------------------------------------------
// Kernel 2: per-state masked GEMM with f32 WMMA.
// grid.x = NUM_STATES * MAX_ROW_TILES (slot -> state, tile; uniform early exit)
// grid.y = N_OUT / (WAVES*16) = 16 column blocks of 64.
// ---------------------------------------------------------------------------
__global__ __launch_bounds__(BLOCK_THREADS)
void masked_gemm_kernel(const float* __restrict__ x,
                        const float* __restrict__ kernelW,
                        const float* __restrict__ masks,
                        const int*   __restrict__ sortedRows,
                        const int*   __restrict__ groupStart,
                        const int*   __restrict__ groupCount,
                        float* __restrict__ out) {
    const int bx   = blockIdx.x;
    const int s    = bx >> 4;          // / MAX_ROW_TILES
    const int tile = bx & (MAX_ROW_TILES - 1);

    const int cnt = groupCount[s];
    if (tile * TILE_M >= cnt) return;  // uniform per-block: EXEC stays all-1s

    const int start    = groupStart[s];
    const int rowsLeft = cnt - tile * TILE_M;   // 1..16 valid rows in this tile

    __shared__ int   rowIdx[TILE_M];
    __shared__ float Ald[TILE_M][A_STRIDE];

    const int t = threadIdx.x;
    if (t < TILE_M) {
        rowIdx[t] = (t < rowsLeft) ? sortedRows[start + tile * TILE_M + t] : 0;
    }
    __syncthreads();

    const int lane  = t & 31;
    const int wave  = t >> 5;
    const int m     = lane & 15;                 // A row / B column within tile
    const int khalf = (lane < 16) ? 0 : 2;       // wave32 A/B half split
    const int ncol  = blockIdx.y * (WAVES * 16) + wave * 16 + m;

    const float* __restrict__ maskS = masks + (size_t)s * K_IN * N_OUT;

    v8f acc = {};

    for (int kc = 0; kc < K_IN; kc += KC) {
        __syncthreads();   // protect Ald from previous chunk's readers

        // Stage 16 gathered x rows x KC floats into LDS (float4, coalesced per row).
        #pragma unroll
        for (int i = t; i < TILE_M * (KC / 4); i += BLOCK_THREADS) {
            const int r  = i / (KC / 4);
            const int c4 = i % (KC / 4);
            const v4f v = *(const v4f*)(x + (size_t)rowIdx[r] * K_IN + kc + c4 * 4);
            *(v4f*)&Ald[r][c4 * 4] = v;          // row stride 1040 B: 16B aligned
        }
        __syncthreads();

        for (int kk = 0; kk < KC; kk += 4) {
            // A fragment: lane holds A[m][k0], A[m][k0+1]  (ds_load_b64, no conflicts)
            const v2f a = *(const v2f*)&Ald[m][kk + khalf];

            // B fragment: lane holds (W .* M)[k0][ncol], (W .* M)[k0+1][ncol]
            const int k0 = kc + kk + khalf;
            v2f b;
            b.x = kernelW[(size_t)k0 * N_OUT + ncol]       * maskS[(size_t)k0 * N_OUT + ncol];
            b.y = kernelW[(size_t)(k0 + 1) * N_OUT + ncol] * maskS[(size_t)(k0 + 1) * N_OUT + ncol];

            // D(16x16,f32) += A(16x4,f32) x B(4x16,f32)
            acc = __builtin_amdgcn_wmma_f32_16x16x4_f32(
                /*neg_a=*/false, a, /*neg_b=*/false, b,
                /*c_mod=*/(short)0, acc, /*reuse_a=*/false, /*reuse_b=*/false);
        }
    }

    // D layout: VGPR j -> M = j + (lane<16 ? 0 : 8), N = lane&15. ReLU + store.
    const int mbase = (lane < 16) ? 0 : 8;
    #pragma unroll
    for (int j = 0; j < 8; ++j) {
        const int lm = mbase + j;
        if (lm < rowsLeft) {
            const int row = rowIdx[lm];
            const float v = acc[j];
            out[(size_t)row * N_OUT + ncol] = (v > 0.0f) ? v : 0.0f;
        }
    }
}

// ---------------------------------------------------------------------------
// Host entry. Inputs (setup_inputs order): x[256*1024] f32, state[256] i32,
// kernel[1024*1024] f32, masks[20*1024*1024] f32. Output: f32 [256*1024].
// ---------------------------------------------------------------------------
extern "C" void kernel_launch(void* const* d_in, const int* in_sizes, int n_in,
                              void* d_out, int out_size, void* d_ws, size_t ws_size,
                              hipStream_t stream) {
    const float* x       = (const float*)d_in[0];
    const int*   state   = (const int*)d_in[1];
    const float* kernelW = (const float*)d_in[2];
    const float* masks   = (const float*)d_in[3];
    float*       out     = (float*)d_out;

    int* sortedRows = (int*)d_ws;                 // 256 ints
    int* groupStart = sortedRows + BS;            // 20 ints
    int* groupCount = groupStart + NUM_STATES;    // 20 ints

    sort_rows_kernel<<<1, BS, 0, stream>>>(state, sortedRows, groupStart, groupCount);

    dim3 grid(NUM_STATES * MAX_ROW_TILES, N_OUT / (WAVES * 16));
    dim3 block(BLOCK_THREADS);
    masked_gemm_kernel<<<grid, block, 0, stream>>>(
        x, kernelW, masks, sortedRows, groupStart, groupCount, out);
}